// BilateralFilter3d_21157008900524
// MI455X (gfx1250) — compile-verified
//
#include <hip/hip_runtime.h>
#include <stdint.h>

// 3D bilateral filter, 5x5x5 window, zero-padded-with-mask semantics.
// Transcendental/VALU-bound (~125 v_exp_f32 + ~550 VALU per voxel; HBM is
// ~100x away from the limit). One v_exp_f32 per tap via log2-domain fusion of
// spatial and range Gaussians. Halo tiles staged to LDS with the CDNA5 async
// global->LDS data mover. Interior blocks (67%) take a mask-free fast path.

#define R   2
#define TX  8
#define TY  8
#define TZ  8
#define HX  (TX + 2*R)        // 12
#define HY  (TY + 2*R)        // 12
#define HZ  (TZ + 2*R)        // 12
#define NT  (HX*HY*HZ)        // 1728 floats = 6912 B of LDS
#define NB  (TX*TY*TZ)        // 512 threads = 16 waves (wave32)
#define N   128
#define NVOX (N*N*N)

typedef __attribute__((address_space(3))) int lds_i32_t;
typedef __attribute__((address_space(1))) int glb_i32_t;

template <bool MASKED>
__device__ __forceinline__ float bilateral_taps(const float* tile, int cidx, float xi,
                                                int gx, int gy, int gz,
                                                float ncx, float ncy, float ncz, float ncc,
                                                float& den_out)
{
    float num = 0.0f, den = 0.0f;
#pragma unroll
    for (int dz = -R; dz <= R; ++dz) {
        bool okz  = !MASKED || ((unsigned)(gz + dz) < (unsigned)N);
        float spz = (float)(dz * dz) * ncz;
#pragma unroll
        for (int dy = -R; dy <= R; ++dy) {
            bool okzy  = !MASKED || (okz && ((unsigned)(gy + dy) < (unsigned)N));
            float spzy = fmaf((float)(dy * dy), ncy, spz);
            int base   = cidx + dz * (HY * HX) + dy * HX;
#pragma unroll
            for (int dx = -R; dx <= R; ++dx) {
                float xk = tile[base + dx];
                float sp = fmaf((float)(dx * dx), ncx, spzy);
                float d  = xi - xk;
                float t  = fmaf(d * d, ncc, sp);      // log2(w_sp * w_range)
                float w  = __builtin_amdgcn_exp2f(t);
                if (MASKED) {
                    bool ok = okzy && ((unsigned)(gx + dx) < (unsigned)N);
                    w = ok ? w : 0.0f;
                }
                num = fmaf(w, xk, num);
                den += w;
            }
        }
    }
    den_out = den;
    return num;
}

__global__ __launch_bounds__(NB)
void bilateral3d_gfx1250_kernel(const float* __restrict__ img,
                                const float* __restrict__ p_sx,
                                const float* __restrict__ p_sy,
                                const float* __restrict__ p_sz,
                                const float* __restrict__ p_cs,
                                float* __restrict__ out)
{
    __shared__ float tile[NT];

    const int tx = threadIdx.x, ty = threadIdx.y, tz = threadIdx.z;
    const int tid = tx + TX * (ty + TY * tz);

    const int bz    = blockIdx.z;        // 0..31 = batch*16 + z-tile
    const int batch = bz >> 4;
    const int x0 = blockIdx.x * TX;
    const int y0 = blockIdx.y * TY;
    const int z0 = (bz & 15) * TZ;

    const float* src = img + (size_t)batch * NVOX;

    // Block is "interior" iff the whole 12^3 halo lies inside the volume.
    const bool interior = (x0 >= R) && (x0 + TX + R <= N) &&
                          (y0 >= R) && (y0 + TY + R <= N) &&
                          (z0 >= R) && (z0 + TZ + R <= N);

    // ---- stage 12x12x12 halo tile into LDS (async DMA, no VGPR roundtrip) ----
    for (int i = tid; i < NT; i += NB) {
        int lz = i / (HY * HX);
        int r  = i - lz * (HY * HX);
        int ly = r / HX;
        int lx = r - ly * HX;
        int gz = z0 + lz - R, gy = y0 + ly - R, gx = x0 + lx - R;
        bool inb = ((unsigned)gz < (unsigned)N) &
                   ((unsigned)gy < (unsigned)N) &
                   ((unsigned)gx < (unsigned)N);
        if (inb) {
            const float* gp = src + ((size_t)gz * N + gy) * N + gx;
#if __has_builtin(__builtin_amdgcn_global_load_async_to_lds_b32)
            __builtin_amdgcn_global_load_async_to_lds_b32(
                (glb_i32_t*)gp, (lds_i32_t*)&tile[i], 0, 0);
#else
            tile[i] = *gp;
#endif
        } else {
            tile[i] = 0.0f;   // masked taps never contribute; zero is safe filler
        }
    }
#if __has_builtin(__builtin_amdgcn_s_wait_asynccnt)
    __builtin_amdgcn_s_wait_asynccnt(0);
#else
    asm volatile("s_wait_asynccnt 0" ::: "memory");
#endif
    __syncthreads();

    // ---- per-thread 125-tap filter, single exp2 per tap ----
    const float L2E = 1.44269504088896340736f;
    const float sx = *p_sx, sy = *p_sy, sz = *p_sz, cs = *p_cs;
    const float ncx = -L2E / (2.0f * sx * sx);
    const float ncy = -L2E / (2.0f * sy * sy);
    const float ncz = -L2E / (2.0f * sz * sz);
    const float ncc = -L2E / (2.0f * cs * cs);

    const int gx = x0 + tx, gy = y0 + ty, gz = z0 + tz;
    const int cidx = ((tz + R) * HY + (ty + R)) * HX + (tx + R);
    const float xi = tile[cidx];

    float num, den;
    if (interior) {   // uniform branch: whole block takes one side
        num = bilateral_taps<false>(tile, cidx, xi, gx, gy, gz, ncx, ncy, ncz, ncc, den);
    } else {
        num = bilateral_taps<true >(tile, cidx, xi, gx, gy, gz, ncx, ncy, ncz, ncc, den);
    }

    out[(size_t)batch * NVOX + ((size_t)gz * N + gy) * N + gx] = num / den;
}

extern "C" void kernel_launch(void* const* d_in, const int* in_sizes, int n_in,
                              void* d_out, int out_size, void* d_ws, size_t ws_size,
                              hipStream_t stream) {
    (void)in_sizes; (void)n_in; (void)out_size; (void)d_ws; (void)ws_size;
    const float* img  = (const float*)d_in[0];
    const float* psx  = (const float*)d_in[1];
    const float* psy  = (const float*)d_in[2];
    const float* psz  = (const float*)d_in[3];
    const float* pcs  = (const float*)d_in[4];
    float*       outp = (float*)d_out;

    dim3 block(TX, TY, TZ);                       // 512 threads = 16 wave32
    dim3 grid(N / TX, N / TY, (N / TZ) * 2);      // 16 x 16 x 32 (batch folded in z)
    bilateral3d_gfx1250_kernel<<<grid, block, 0, stream>>>(img, psx, psy, psz, pcs, outp);
}